// MGF_55860344651876
// MI455X (gfx1250) — compile-verified
//
#include <hip/hip_runtime.h>

typedef __attribute__((ext_vector_type(2))) float v2f;
typedef __attribute__((ext_vector_type(8))) float v8f;

#define TILE 64
#define HALO 18          // top/left halo = k_max + 1 (SAT needs index y-k-1)
#define REG  99          // loaded region: 64 + 18 + 17
#define NB   8           // 8 blocks of 16 -> 128x128 padded SAT workspace
#define SW   129         // LDS row stride in floats (odd -> conflict-free col walks)
#define IMH  512
#define IMW  512

__device__ __forceinline__ int reflect512(int i) {
    i = (i < 0) ? -i : i;
    i = (i > 511) ? (1022 - i) : i;
    return i;
}

__global__ __launch_bounds__(256)
void mgf_sat_wmma_kernel(const float* __restrict__ x, float* __restrict__ out) {
    extern __shared__ float S[];            // [128][SW], SAT workspace
    const int tid  = threadIdx.x;
    const int lane = tid & 31;
    const int wave = tid >> 5;
    const int tx0  = blockIdx.x * TILE;
    const int ty0  = blockIdx.y * TILE;
    const int b    = blockIdx.z;
    const float* img = x + (size_t)b * IMH * IMW;

    // ---- zero the 128x128 workspace (zero padding is prefix-sum-neutral) ----
    for (int i = tid; i < 128 * SW; i += 256) S[i] = 0.0f;
    __syncthreads();

    // ---- load reflect-padded region (+0.01); one halo serves all 3 scales ----
    for (int i = tid; i < REG * REG; i += 256) {
        int r  = i / REG, c = i % REG;
        int gy = reflect512(ty0 - HALO + r);
        int gx = reflect512(tx0 - HALO + c);
        S[r * SW + c] = img[gy * IMW + gx] + 0.01f;
    }
    __syncthreads();

    const int lr = lane & 15;   // A-row / B-col index within 16x16 block
    const int hi = lane >> 4;   // half-wave selector (K/M split per ISA layout)

    // ==== row-prefix pass: per 16x16 block, Y = X * U  (U upper-tri ones) ====
    // wave w owns block-row w; sequential over block-cols with scalar carry.
    {
        const int i0 = wave * 16;
        for (int j = 0; j < NB; ++j) {
            const int c0 = j * 16;
            v8f acc = {};
            #pragma unroll
            for (int q = 0; q < 4; ++q) {
                // A chunk (16x4 of X): lane row = lr, cols = q*4 + 2*hi + {0,1}
                v2f a;
                a.x = S[(i0 + lr) * SW + c0 + q * 4 + 2 * hi + 0];
                a.y = S[(i0 + lr) * SW + c0 + q * 4 + 2 * hi + 1];
                // B chunk (4x16 of U): U[r][c] = (c >= r)
                const int r0 = q * 4 + 2 * hi;
                v2f bm;
                bm.x = (lr >= r0)     ? 1.0f : 0.0f;
                bm.y = (lr >= r0 + 1) ? 1.0f : 0.0f;
                acc = __builtin_amdgcn_wmma_f32_16x16x4_f32(
                        false, a, false, bm, (short)0, acc, false, false);
            }
            if (j > 0) {                       // carry = running row prefix
                #pragma unroll
                for (int v = 0; v < 8; ++v)
                    acc[v] += S[(i0 + v + hi * 8) * SW + c0 - 1];
            }
            #pragma unroll
            for (int v = 0; v < 8; ++v)        // D layout: row v + 8*hi, col lr
                S[(i0 + v + hi * 8) * SW + c0 + lr] = acc[v];
        }
    }
    __syncthreads();

    // ==== col-prefix pass: per 16x16 block, Z = L * Y  (L lower-tri ones) ====
    // wave w owns block-col w; sequential over block-rows with scalar carry.
    {
        const int j0 = wave * 16;
        for (int i = 0; i < NB; ++i) {
            const int r0b = i * 16;
            v8f acc = {};
            #pragma unroll
            for (int q = 0; q < 4; ++q) {
                const int k0 = q * 4 + 2 * hi;
                // A chunk (16x4 of L): L[m][k] = (k <= m)
                v2f a;
                a.x = (k0     <= lr) ? 1.0f : 0.0f;
                a.y = (k0 + 1 <= lr) ? 1.0f : 0.0f;
                // B chunk (4x16 of Y): rows k0+{0,1}, col = j0 + lr
                v2f bm;
                bm.x = S[(r0b + k0 + 0) * SW + j0 + lr];
                bm.y = S[(r0b + k0 + 1) * SW + j0 + lr];
                acc = __builtin_amdgcn_wmma_f32_16x16x4_f32(
                        false, a, false, bm, (short)0, acc, false, false);
            }
            if (i > 0) {                       // carry = column prefix so far
                const float cr = S[(r0b - 1) * SW + j0 + lr];
                #pragma unroll
                for (int v = 0; v < 8; ++v) acc[v] += cr;
            }
            #pragma unroll
            for (int v = 0; v < 8; ++v)
                S[(r0b + v + hi * 8) * SW + j0 + lr] = acc[v];
        }
    }
    __syncthreads();

    // ==== output: 12 SAT lookups per (pixel, scale) -> log-ratio magnitude ====
    const int kk[3] = {9, 13, 17};
    for (int p = tid; p < TILE * TILE; p += 256) {
        const int py = p >> 6, px = p & 63;
        const int gy = ty0 + py, gx = tx0 + px;
        const int r0 = py + HALO, c0 = px + HALO;
        #pragma unroll
        for (int s = 0; s < 3; ++s) {
            const int k = kk[s];
            const float s_pp = S[(r0 + k) * SW + (c0 + k)];
            const float s_mp = S[(r0 - k - 1) * SW + (c0 + k)];
            const float s_pm = S[(r0 + k) * SW + (c0 - k - 1)];
            const float s_mm = S[(r0 - k - 1) * SW + (c0 - k - 1)];
            const float s_p0 = S[(r0 + k) * SW + c0];
            const float s_m0 = S[(r0 - k - 1) * SW + c0];
            const float s_0p = S[r0 * SW + (c0 + k)];
            const float s_0m = S[r0 * SW + (c0 - k - 1)];
            const float s_pL = S[(r0 + k) * SW + (c0 - 1)];
            const float s_mL = S[(r0 - k - 1) * SW + (c0 - 1)];
            const float s_Tp = S[(r0 - 1) * SW + (c0 + k)];
            const float s_Tm = S[(r0 - 1) * SW + (c0 - k - 1)];
            const float gl = s_p0 - s_m0 - s_pm + s_mm;   // rows +-k, cols [-k,0]
            const float gr = s_pp - s_mp - s_pL + s_mL;   // rows +-k, cols [0,k]
            const float gt = s_0p - s_mp - s_0m + s_mm;   // rows [-k,0], cols +-k
            const float gb = s_pp - s_Tp - s_pm + s_Tm;   // rows [0,k],  cols +-k
            const float gxv = logf(gl / (gr + 1e-8f));
            const float gyv = logf(gt / (gb + 1e-8f));
            out[(((size_t)b * 3 + s) * IMH + gy) * IMW + gx] =
                sqrtf(gxv * gxv + gyv * gyv);
        }
    }
}

extern "C" void kernel_launch(void* const* d_in, const int* in_sizes, int n_in,
                              void* d_out, int out_size, void* d_ws, size_t ws_size,
                              hipStream_t stream) {
    const float* x = (const float*)d_in[0];
    float* out = (float*)d_out;
    const int Bimg = in_sizes[0] / (IMH * IMW);
    dim3 grid(IMW / TILE, IMH / TILE, Bimg);
    const size_t shmem = (size_t)128 * SW * sizeof(float);
    mgf_sat_wmma_kernel<<<grid, 256, shmem, stream>>>(x, out);
}